// EdgeMLP_69398081569224
// MI455X (gfx1250) — compile-verified
//
#include <hip/hip_runtime.h>
#include <stdint.h>

#define N_ROWS 1024
#define ZD     64
#define HID    128
#define TI     64
#define TJ     64
#define LDP    132   // LDS row pitch (floats): 16B-aligned rows, gcd(LDP,64)=4 -> conflict-free strided reads

typedef __attribute__((ext_vector_type(2))) float v2f;
typedef __attribute__((ext_vector_type(4))) float v4f;
typedef __attribute__((ext_vector_type(8))) float v8f;

// ---------------------------------------------------------------------------
// Kernel A: pi = z @ Wi^T, pj = z @ Wj^T + b1   via V_WMMA_F32_16X16X4_F32
// One wave per 16x16 output tile of BOTH pi and pj (A operand reused).
// 512 waves total = 64 blocks x 8 waves.
// ---------------------------------------------------------------------------
__global__ __launch_bounds__(256) void proj_wmma_kernel(
    const float* __restrict__ z,   // 1024 x 64
    const float* __restrict__ W1,  // 128 x 128  (Wi = cols 0..63, Wj = cols 64..127)
    const float* __restrict__ b1,  // 128
    float* __restrict__ pi,        // 1024 x 128
    float* __restrict__ pj)        // 1024 x 128  (includes +b1)
{
  const int lane = threadIdx.x & 31;
  const int wave = blockIdx.x * 8 + (threadIdx.x >> 5); // 0..511
  const int tm = wave >> 3;                 // z-row tile    0..63
  const int tn = wave & 7;                  // hid-col tile  0..7

  // A (16x4 f32): lanes 0-15 hold M=0..15 K={0,1}; lanes 16-31 hold K={2,3}
  const int zrow = tm * 16 + (lane & 15);
  const int wrow = tn * 16 + (lane & 15);   // W1 row == output hid column
  const int kh   = (lane >> 4) << 1;        // 0 or 2

  v8f acc_i = {};
  v8f acc_j = {};
#pragma unroll
  for (int k0 = 0; k0 < ZD; k0 += 4) {
    v2f a  = *(const v2f*)(z  + zrow * ZD        + k0 + kh);
    v2f bi = *(const v2f*)(W1 + wrow * (2 * ZD)       + k0 + kh);
    v2f bj = *(const v2f*)(W1 + wrow * (2 * ZD) + ZD  + k0 + kh);
    acc_i = __builtin_amdgcn_wmma_f32_16x16x4_f32(
        /*neg_a=*/false, a, /*neg_b=*/false, bi,
        /*c_mod=*/(short)0, acc_i, /*reuse_a=*/false, /*reuse_b=*/false);
    acc_j = __builtin_amdgcn_wmma_f32_16x16x4_f32(
        false, a, false, bj, (short)0, acc_j, false, false);
  }

  // D layout: vgpr r -> row (r or r+8), lanes 0-15 cols 0-15, lanes 16-31 rows +8
  const int ocol  = tn * 16 + (lane & 15);
  const int rbase = tm * 16 + ((lane >> 4) << 3);
  const float b1v = b1[ocol];
#pragma unroll
  for (int r = 0; r < 8; ++r) {
    pi[(rbase + r) * HID + ocol] = acc_i[r];
    pj[(rbase + r) * HID + ocol] = acc_j[r] + b1v;
  }
}

// ---------------------------------------------------------------------------
// Kernel B: out[i,j] = b2 + sum_h relu(pi[i,h] + pj[j,h]) * w2[h]
// 64x64 tile per block, 256 threads, 4x4 micro-tile per thread (strided).
// Tiles staged into LDS with GLOBAL_LOAD_ASYNC_TO_LDS_B128 (ASYNCcnt path,
// no VGPR staging), GVS addressing: SGPR base + 32-bit VGPR byte offset.
// ---------------------------------------------------------------------------
__global__ __launch_bounds__(256) void edge_score_kernel(
    const float* __restrict__ pi,
    const float* __restrict__ pj,
    const float* __restrict__ w2,  // 128
    const float* __restrict__ b2,  // 1
    float* __restrict__ out)       // 1024 x 1024
{
  __shared__ float spi[TI * LDP];
  __shared__ float spj[TJ * LDP];
  __shared__ float sw[HID];

  const int tid = threadIdx.x;
  const int i0g = blockIdx.y * TI;
  const int j0g = blockIdx.x * TJ;

  // Low 32 bits of the shared-aperture generic address == LDS byte offset.
  const uint32_t spi_base = (uint32_t)(uintptr_t)(&spi[0]);
  const uint32_t spj_base = (uint32_t)(uintptr_t)(&spj[0]);

  // Async stage: 64 rows x 128 cols per tile, 16B per lane, 8 iters/thread.
#pragma unroll
  for (int t = tid; t < TI * (HID / 4); t += 256) {
    const int r  = t >> 5;            // HID/4 == 32 float4 chunks per row
    const int c4 = (t & 31) << 2;
    const uint32_t loff   = (uint32_t)((r * LDP + c4) * 4);
    const uint32_t goff_i = (uint32_t)(((i0g + r) * HID + c4) * 4);
    const uint32_t goff_j = (uint32_t)(((j0g + r) * HID + c4) * 4);
    asm volatile("global_load_async_to_lds_b128 %0, %1, %2"
                 :: "v"(spi_base + loff), "v"(goff_i), "s"(pi)
                 : "memory");
    asm volatile("global_load_async_to_lds_b128 %0, %1, %2"
                 :: "v"(spj_base + loff), "v"(goff_j), "s"(pj)
                 : "memory");
  }
  if (tid < HID) sw[tid] = w2[tid];

  asm volatile("s_wait_asynccnt 0x0" ::: "memory");
  __syncthreads();

  const int tj = tid & 15;   // j micro index (lane-contiguous for coalesced stores)
  const int ti = tid >> 4;   // i micro index

  float acc[4][4] = {};
#pragma unroll 4
  for (int h = 0; h < HID; ++h) {
    const float w = sw[h];
    float a[4], b[4];
#pragma unroll
    for (int r = 0; r < 4; ++r) a[r] = spi[(ti + 16 * r) * LDP + h];
#pragma unroll
    for (int c = 0; c < 4; ++c) b[c] = spj[(tj + 16 * c) * LDP + h];
#pragma unroll
    for (int r = 0; r < 4; ++r)
#pragma unroll
      for (int c = 0; c < 4; ++c) {
        float x = a[r] + b[c];
        x = fmaxf(x, 0.0f);
        acc[r][c] = __builtin_fmaf(x, w, acc[r][c]);
      }
  }

  const float bias = b2[0];
#pragma unroll
  for (int r = 0; r < 4; ++r)
#pragma unroll
    for (int c = 0; c < 4; ++c) {
      out[(size_t)(i0g + ti + 16 * r) * N_ROWS + (j0g + tj + 16 * c)] =
          acc[r][c] + bias;
    }
}

// ---------------------------------------------------------------------------
extern "C" void kernel_launch(void* const* d_in, const int* in_sizes, int n_in,
                              void* d_out, int out_size, void* d_ws, size_t ws_size,
                              hipStream_t stream) {
  const float* z  = (const float*)d_in[0];
  const float* W1 = (const float*)d_in[1];
  const float* b1 = (const float*)d_in[2];
  const float* W2 = (const float*)d_in[3];
  const float* b2 = (const float*)d_in[4];
  float* out = (float*)d_out;

  float* pi = (float*)d_ws;            // 1024*128 floats
  float* pj = pi + N_ROWS * HID;       // 1024*128 floats (total 1 MB of ws)

  proj_wmma_kernel<<<dim3(64), dim3(256), 0, stream>>>(z, W1, b1, pi, pj);
  edge_score_kernel<<<dim3(N_ROWS / TJ, N_ROWS / TI), dim3(256), 0, stream>>>(
      pi, pj, W2, b2, out);
}